// TransitionDown_71820443124432
// MI455X (gfx1250) — compile-verified
//
#include <hip/hip_runtime.h>

#define NB    8
#define NPTS  4096
#define DIN   128
#define DOUT  256
#define KNN   16
#define NSAMP 1024
#define NROWS (NB * NSAMP * KNN)   /* 131072 */
#define BN_EPS 1e-5f

typedef float v2f __attribute__((ext_vector_type(2)));
typedef float v8f __attribute__((ext_vector_type(8)));
typedef unsigned long long u64;

// ============================================================================
// Kernel 1: Furthest point sampling. One block per batch, 1024 threads,
// each thread owns 4 points (coords + running min-dist in registers).
// Per step: distance update, wave32 shuffle argmax, 32-entry LDS argmax.
// Packed u64 key (dist_bits<<32 | (4095-p)) gives argmax with
// smallest-index tie-break, matching jnp.argmax.
// ============================================================================
__global__ __launch_bounds__(1024) void fps_kernel(const float* __restrict__ xyz,
                                                   int* __restrict__ sample_idx,
                                                   float* __restrict__ sample_xyz)
{
    __shared__ float sx[NPTS], sy[NPTS], sz[NPTS];
    __shared__ u64 red[32];
    __shared__ float cur[3];

    const int b = blockIdx.x;
    const int t = threadIdx.x;
    const float* base = xyz + (size_t)b * NPTS * 3;

    float px[4], py[4], pz[4], dist[4];
#pragma unroll
    for (int j = 0; j < 4; ++j) {
        int p = t + j * 1024;
        float x = base[p * 3 + 0];
        float y = base[p * 3 + 1];
        float z = base[p * 3 + 2];
        sx[p] = x; sy[p] = y; sz[p] = z;
        px[j] = x; py[j] = y; pz[j] = z;
        dist[j] = 1.0e10f;
    }
    if (t == 0) {
        cur[0] = base[0]; cur[1] = base[1]; cur[2] = base[2];
        sample_idx[b * NSAMP] = 0;
        float* o3 = sample_xyz + (size_t)b * NSAMP * 3;
        o3[0] = base[0]; o3[1] = base[1]; o3[2] = base[2];
    }
    __syncthreads();

    for (int i = 1; i < NSAMP; ++i) {
        float cx = cur[0], cy = cur[1], cz = cur[2];
        u64 best = 0ull;
#pragma unroll
        for (int j = 0; j < 4; ++j) {
            float dx = px[j] - cx, dy = py[j] - cy, dz = pz[j] - cz;
            float d = dx * dx + dy * dy + dz * dz;
            dist[j] = fminf(dist[j], d);
            int p = t + j * 1024;
            u64 key = ((u64)__float_as_uint(dist[j]) << 32) | (unsigned)(4095 - p);
            best = (key > best) ? key : best;
        }
#pragma unroll
        for (int o = 16; o > 0; o >>= 1) {
            u64 other = __shfl_xor(best, o);
            best = (other > best) ? other : best;
        }
        __syncthreads();
        if ((t & 31) == 0) red[t >> 5] = best;
        __syncthreads();
        if (t < 32) {
            u64 v = red[t];
#pragma unroll
            for (int o = 16; o > 0; o >>= 1) {
                u64 other = __shfl_xor(v, o);
                v = (other > v) ? other : v;
            }
            if (t == 0) {
                int p = 4095 - (int)(v & 0xffffffffull);
                cur[0] = sx[p]; cur[1] = sy[p]; cur[2] = sz[p];
                sample_idx[b * NSAMP + i] = p;
                float* o3 = sample_xyz + ((size_t)b * NSAMP + i) * 3;
                o3[0] = sx[p]; o3[1] = sy[p]; o3[2] = sz[p];
            }
        }
        __syncthreads();
    }
}

// ============================================================================
// Kernel 2: KNN (16 nearest of 4096 per sample). One wave32 per sample.
// Per-lane sorted top-16 over 128 points, then 32-way merge: per-lane
// cursor into its sorted LDS list + wave-min over packed (dist,idx) keys.
// ============================================================================
__global__ __launch_bounds__(256) void knn_kernel(const float* __restrict__ xyz,
                                                  const float* __restrict__ sample_xyz,
                                                  int* __restrict__ knn_idx)
{
    __shared__ u64 cand[8 * 32 * 16];   // 8 waves * 512 entries = 32 KB
    const int wave = threadIdx.x >> 5;
    const int lane = threadIdx.x & 31;
    const int gs = blockIdx.x * 8 + wave;          // sample id [0, 8192)
    const int b = gs >> 10;
    const float* q3 = sample_xyz + (size_t)gs * 3;
    const float qx = q3[0], qy = q3[1], qz = q3[2];
    const float* base = xyz + (size_t)b * NPTS * 3;

    float bd[KNN];
    int bi[KNN];
#pragma unroll
    for (int j = 0; j < KNN; ++j) { bd[j] = 3.0e38f; bi[j] = 0; }

    for (int p = lane; p < NPTS; p += 32) {
        float dx = base[p * 3 + 0] - qx;
        float dy = base[p * 3 + 1] - qy;
        float dz = base[p * 3 + 2] - qz;
        float d = dx * dx + dy * dy + dz * dz;
        if (d < bd[KNN - 1]) {
            bd[KNN - 1] = d; bi[KNN - 1] = p;
#pragma unroll
            for (int j = KNN - 1; j > 0; --j) {
                if (bd[j] < bd[j - 1]) {
                    float td = bd[j]; bd[j] = bd[j - 1]; bd[j - 1] = td;
                    int ti = bi[j]; bi[j] = bi[j - 1]; bi[j - 1] = ti;
                }
            }
        }
    }

    u64* wc = &cand[(wave * 32 + lane) * 16];
#pragma unroll
    for (int j = 0; j < KNN; ++j)
        wc[j] = ((u64)__float_as_uint(bd[j]) << 32) | (unsigned)bi[j];
    __syncthreads();

    u64* wbase = &cand[wave * 512];
    int curp = 0;
    int* out = knn_idx + (size_t)gs * KNN;
    for (int r = 0; r < KNN; ++r) {
        u64 c = wbase[lane * 16 + curp];
        u64 mn = c;
#pragma unroll
        for (int o = 16; o > 0; o >>= 1) {
            u64 other = __shfl_xor(mn, o);
            mn = (other < mn) ? other : mn;
        }
        if (c == mn) curp++;
        if (lane == 0) out[r] = (int)(mn & 0xffffffffull);
    }
}

// ============================================================================
// Kernel 3/5: gathered GEMM via V_WMMA_F32_16X16X4_F32.
// Block = 512 threads = 16 waves; wave w owns channels [16w, 16w+16).
// B fragments (W columns) register-resident for the whole kernel (64 VGPRs).
// Each (b,s) tile = its 16 neighbors x 128 feats, staged in LDS (stride 132).
// mode 0: accumulate per-channel sum / sumsq (BN stats) via atomics.
// mode 1: BN + ReLU + max over the 16 rows (= max over k), write output.
// ============================================================================
__global__ __launch_bounds__(512) void gemm_pass(const float* __restrict__ feat,
                                                 const float* __restrict__ Wm,
                                                 const float* __restrict__ bias,
                                                 const float* __restrict__ gamma,
                                                 const float* __restrict__ beta,
                                                 const int* __restrict__ knn_idx,
                                                 const float* __restrict__ meanv,
                                                 const float* __restrict__ invstdv,
                                                 float* __restrict__ stat_sum,
                                                 float* __restrict__ stat_sq,
                                                 float* __restrict__ outF,
                                                 int mode)
{
    __shared__ float As[16 * 132];   // 16 rows x 128, stride 132 (bank-safe, 16B-aligned)
    const int lane = threadIdx.x & 31;
    const int wave = threadIdx.x >> 5;
    const int m    = lane & 15;
    const int hi   = lane >> 4;          // 0: K{0,1}/M rows 0-7, 1: K{2,3}/M rows 8-15
    const int n    = wave * 16 + m;      // this lane's output channel

    // Preload this lane's B fragments: W[n][4kk + 2*hi], W[n][4kk + 2*hi + 1]
    v2f bfrag[32];
    const float* wrow = Wm + (size_t)n * DIN;
#pragma unroll
    for (int kk = 0; kk < 32; ++kk) {
        float4 q = ((const float4*)wrow)[kk];
        v2f bf;
        bf.x = hi ? q.z : q.x;
        bf.y = hi ? q.w : q.y;
        bfrag[kk] = bf;
    }
    const float bias_n = bias[n];
    float gscale = 0.f, gshift = 0.f;
    if (mode == 1) {
        float g = gamma[n] * invstdv[n];
        gscale = g;
        gshift = beta[n] - g * meanv[n];
    }

    for (int t = blockIdx.x; t < NB * NSAMP; t += gridDim.x) {
        // Stage gathered A tile: 512 threads, one float4 each (16 x 128)
        {
            int r = threadIdx.x >> 5;            // 0..15  (row = neighbor k)
            int c = (threadIdx.x & 31) * 4;      // 0..124
            int b = t >> 10;
            int idx = knn_idx[(size_t)t * KNN + r];
            const float* rowp = feat + ((size_t)(b * NPTS + idx)) * DIN;
            *(float4*)&As[r * 132 + c] = *(const float4*)(rowp + c);
        }
        __syncthreads();

        v8f acc = {0.f, 0.f, 0.f, 0.f, 0.f, 0.f, 0.f, 0.f};
        const float* arow = &As[m * 132 + hi * 2];
#pragma unroll
        for (int kk = 0; kk < 32; ++kk) {
            v2f a = *(const v2f*)(arow + kk * 4);
            acc = __builtin_amdgcn_wmma_f32_16x16x4_f32(
                false, a, false, bfrag[kk], (short)0, acc, false, false);
        }
        __syncthreads();

        if (mode == 0) {
            float s = 0.f, ss = 0.f;
#pragma unroll
            for (int j = 0; j < 8; ++j) {
                float h = acc[j] + bias_n;
                s += h; ss += h * h;
            }
            s  += __shfl_down(s, 16);
            ss += __shfl_down(ss, 16);
            if (hi == 0) {
                atomicAdd(&stat_sum[n], s);
                atomicAdd(&stat_sq[n], ss);
            }
        } else {
            float mx = 0.f;   // ReLU output is >= 0
#pragma unroll
            for (int j = 0; j < 8; ++j) {
                float h = (acc[j] + bias_n) * gscale + gshift;
                mx = fmaxf(mx, fmaxf(h, 0.f));
            }
            mx = fmaxf(mx, __shfl_down(mx, 16));
            if (hi == 0) outF[(size_t)t * DOUT + n] = mx;
        }
    }
}

// ============================================================================
// Stats helpers
// ============================================================================
__global__ void zero_stats(float* s) {
    int i = blockIdx.x * blockDim.x + threadIdx.x;
    if (i < 2 * DOUT) s[i] = 0.f;
}

__global__ void finalize_stats(const float* __restrict__ ssum,
                               const float* __restrict__ ssq,
                               float* __restrict__ meanv,
                               float* __restrict__ invstdv)
{
    int n = threadIdx.x;
    if (n < DOUT) {
        const float inv_cnt = 1.0f / (float)NROWS;
        float mean = ssum[n] * inv_cnt;
        float var = ssq[n] * inv_cnt - mean * mean;
        meanv[n] = mean;
        invstdv[n] = rsqrtf(var + BN_EPS);
    }
}

// ============================================================================
// Launch
// ============================================================================
extern "C" void kernel_launch(void* const* d_in, const int* in_sizes, int n_in,
                              void* d_out, int out_size, void* d_ws, size_t ws_size,
                              hipStream_t stream)
{
    const float* feat  = (const float*)d_in[0];   // (8,4096,128)
    const float* xyz   = (const float*)d_in[1];   // (8,4096,3)
    const float* Wm    = (const float*)d_in[2];   // (256,128)
    const float* bias  = (const float*)d_in[3];   // (256,)
    const float* gamma = (const float*)d_in[4];   // (256,)
    const float* beta  = (const float*)d_in[5];   // (256,)

    float* out_feat = (float*)d_out;                               // (8,1024,256)
    float* out_xyz  = (float*)d_out + (size_t)NB * NSAMP * DOUT;   // (8,1024,3)

    // workspace layout
    int*   sample_idx = (int*)d_ws;                                  // 8192
    int*   knn_idx    = sample_idx + NB * NSAMP;                     // 131072
    float* ssum       = (float*)(knn_idx + NB * NSAMP * KNN);        // 256
    float* ssq        = ssum + DOUT;                                 // 256
    float* meanv      = ssum + 2 * DOUT;                             // 256
    float* invstdv    = ssum + 3 * DOUT;                             // 256

    fps_kernel<<<NB, 1024, 0, stream>>>(xyz, sample_idx, out_xyz);
    knn_kernel<<<(NB * NSAMP) / 8, 256, 0, stream>>>(xyz, out_xyz, knn_idx);
    zero_stats<<<1, 512, 0, stream>>>(ssum);
    gemm_pass<<<2048, 512, 0, stream>>>(feat, Wm, bias, gamma, beta, knn_idx,
                                        meanv, invstdv, ssum, ssq, out_feat, 0);
    finalize_stats<<<1, 256, 0, stream>>>(ssum, ssq, meanv, invstdv);
    gemm_pass<<<2048, 512, 0, stream>>>(feat, Wm, bias, gamma, beta, knn_idx,
                                        meanv, invstdv, ssum, ssq, out_feat, 1);
}